// SGC_63376537420316
// MI455X (gfx1250) — compile-verified
//
#include <hip/hip_runtime.h>
#include <math.h>

#define N_NODES   50000
#define N_EDGES   1600000
#define IN_FEATS  256
#define HIDDEN    128
#define N_CLASSES 40

typedef float v2f __attribute__((ext_vector_type(2)));
typedef float v8f __attribute__((ext_vector_type(8)));

// ---------------------------------------------------------------- utilities
__global__ void gcn_zero_kernel(float* __restrict__ p, int n) {
    int i = blockIdx.x * blockDim.x + threadIdx.x;
    if (i < n) p[i] = 0.0f;
}

// degree accumulation: one thread per edge
__global__ void gcn_degree_kernel(const int* __restrict__ src,
                                  const int* __restrict__ dst,
                                  float* __restrict__ degO,
                                  float* __restrict__ degI) {
    int e = blockIdx.x * blockDim.x + threadIdx.x;
    if (e < N_EDGES) {
        atomicAdd(&degO[src[e]], 1.0f);
        atomicAdd(&degI[dst[e]], 1.0f);
    }
}

// deg -> deg^{-1/2} (in place), with the reference's where(deg>0, rsqrt(max(deg,1)), 0)
__global__ void gcn_isqrt_kernel(float* __restrict__ degO, float* __restrict__ degI) {
    int i = blockIdx.x * blockDim.x + threadIdx.x;
    if (i < N_NODES) {
        float a = degO[i];
        float b = degI[i];
        degO[i] = (a > 0.0f) ? rsqrtf(fmaxf(a, 1.0f)) : 0.0f;
        degI[i] = (b > 0.0f) ? rsqrtf(fmaxf(b, 1.0f)) : 0.0f;
    }
}

// ------------------------------------------------- WMMA f32 GEMM (scaled A)
// Y[row][n] = sum_k (X[row][k] * scale[row]) * W[k][n]
// One wave (32 lanes) per 16-row block; NTILES 16-wide column tiles.
// Uses V_WMMA_F32_16X16X4_F32 (exact f32, matches reference numerics).
template <int K, int NCOLS, int NTILES>
__global__ __launch_bounds__(32)
void gcn_wmma_gemm(const float* __restrict__ X,
                   const float* __restrict__ W,
                   const float* __restrict__ scale,
                   float* __restrict__ Y) {
    const int lane = threadIdx.x & 31;
    const int lr = lane & 15;   // A row / B,C column within tile
    const int lh = lane >> 4;   // half-wave select
    const int rowBase = blockIdx.x * 16;
    const int row = rowBase + lr;

    const float s = scale[row];
    const float* __restrict__ xrow = X + (size_t)row * K;

    v8f acc[NTILES];
#pragma unroll
    for (int t = 0; t < NTILES; ++t) acc[t] = v8f{};

    for (int k0 = 0; k0 < K; k0 += 4) {
        const int ka = k0 + 2 * lh;           // A: lane holds K = ka, ka+1
        v2f a;
        a.x = xrow[ka] * s;                    // contiguous float2 load
        a.y = xrow[ka + 1] * s;
#pragma unroll
        for (int t = 0; t < NTILES; ++t) {
            const int n = t * 16 + lr;
            v2f b;
            b.x = (n < NCOLS) ? W[(size_t)ka * NCOLS + n] : 0.0f;
            b.y = (n < NCOLS) ? W[(size_t)(ka + 1) * NCOLS + n] : 0.0f;
            // (neg_a, A, neg_b, B, c_mod, C, reuse_a, reuse_b)
            acc[t] = __builtin_amdgcn_wmma_f32_16x16x4_f32(
                false, a, false, b, (short)0, acc[t], false, false);
        }
    }

#pragma unroll
    for (int t = 0; t < NTILES; ++t) {
        const int col = t * 16 + lr;
        if (col < NCOLS) {
#pragma unroll
            for (int v = 0; v < 8; ++v) {
                const int r = rowBase + v + 8 * lh;   // C/D layout
                Y[(size_t)r * NCOLS + col] = acc[t][v];
            }
        }
    }
}

// ---------------------------------------------- edge scatter (layer 1, F=128)
// 32 threads per edge, each handles a contiguous float4 of the message row.
__global__ void gcn_scatter128_kernel(const float* __restrict__ h,
                                      const int* __restrict__ src,
                                      const int* __restrict__ dst,
                                      float* __restrict__ agg) {
    const int t = blockIdx.x * blockDim.x + threadIdx.x;   // < N_EDGES*32
    const int e = t >> 5;
    if (e < N_EDGES) {
        const int f0 = (t & 31) * 4;
        const float4 v = *(const float4*)(h + (size_t)src[e] * HIDDEN + f0);
        float* ap = agg + (size_t)dst[e] * HIDDEN + f0;
        atomicAdd(ap + 0, v.x);
        atomicAdd(ap + 1, v.y);
        atomicAdd(ap + 2, v.z);
        atomicAdd(ap + 3, v.w);
    }
}

// ---------------------------------------------- edge scatter (layer 2, F=40)
__global__ void gcn_scatter40_kernel(const float* __restrict__ h2,
                                     const int* __restrict__ src,
                                     const int* __restrict__ dst,
                                     float* __restrict__ agg2) {
    const int t = blockIdx.x * blockDim.x + threadIdx.x;   // < N_EDGES*40
    if (t < N_EDGES * N_CLASSES) {
        const int e = t / N_CLASSES;
        const int c = t - e * N_CLASSES;
        atomicAdd(&agg2[(size_t)dst[e] * N_CLASSES + c],
                  h2[(size_t)src[e] * N_CLASSES + c]);
    }
}

// ------------------------------------ layer-1 normalize + bias (in place)
__global__ void gcn_normbias_kernel(float* __restrict__ agg,
                                    const float* __restrict__ degIisq,
                                    const float* __restrict__ bias) {
    const int i = blockIdx.x * blockDim.x + threadIdx.x;   // < N_NODES*HIDDEN
    if (i < N_NODES * HIDDEN) {
        const int node = i >> 7;
        const int f = i & (HIDDEN - 1);
        agg[i] = agg[i] * degIisq[node] + bias[f];
    }
}

// --------------- layer-2 normalize + bias + log_softmax, one wave per node
__global__ __launch_bounds__(256)
void gcn_lsm_kernel(const float* __restrict__ agg2,
                    const float* __restrict__ degIisq,
                    const float* __restrict__ bias,
                    float* __restrict__ out) {
    const int lane = threadIdx.x & 31;
    const int wave = threadIdx.x >> 5;
    const int node = blockIdx.x * 8 + wave;
    if (node >= N_NODES) return;

    const float s = degIisq[node];
    const float* row = agg2 + (size_t)node * N_CLASSES;

    const int c1 = lane + 32;
    float x0 = (lane < N_CLASSES) ? row[lane] * s + bias[lane] : -INFINITY;
    float x1 = (c1 < N_CLASSES) ? row[c1] * s + bias[c1] : -INFINITY;

    float m = fmaxf(x0, x1);
#pragma unroll
    for (int off = 16; off > 0; off >>= 1)
        m = fmaxf(m, __shfl_xor(m, off, 32));

    float sum = ((lane < N_CLASSES) ? expf(x0 - m) : 0.0f) +
                ((c1 < N_CLASSES) ? expf(x1 - m) : 0.0f);
#pragma unroll
    for (int off = 16; off > 0; off >>= 1)
        sum += __shfl_xor(sum, off, 32);

    const float lse = m + logf(sum);
    if (lane < N_CLASSES) out[(size_t)node * N_CLASSES + lane] = x0 - lse;
    if (c1 < N_CLASSES)   out[(size_t)node * N_CLASSES + c1] = x1 - lse;
}

// ---------------------------------------------------------------- launcher
extern "C" void kernel_launch(void* const* d_in, const int* in_sizes, int n_in,
                              void* d_out, int out_size, void* d_ws, size_t ws_size,
                              hipStream_t stream) {
    (void)in_sizes; (void)n_in; (void)out_size; (void)ws_size;
    const float* x  = (const float*)d_in[0];
    const int* src  = (const int*)d_in[1];
    const int* dst  = (const int*)d_in[2];
    const float* W1 = (const float*)d_in[3];
    const float* b1 = (const float*)d_in[4];
    const float* W2 = (const float*)d_in[5];
    const float* b2 = (const float*)d_in[6];
    float* out = (float*)d_out;

    // workspace layout (floats)
    float* ws   = (float*)d_ws;
    float* degO = ws;                                   // N_NODES  (-> deg_out^-1/2)
    float* degI = degO + N_NODES;                       // N_NODES  (-> deg_in^-1/2)
    float* h    = degI + N_NODES;                       // N_NODES*HIDDEN
    float* agg1 = h + (size_t)N_NODES * HIDDEN;         // N_NODES*HIDDEN
    float* h2   = agg1 + (size_t)N_NODES * HIDDEN;      // N_NODES*N_CLASSES
    float* agg2 = h2 + (size_t)N_NODES * N_CLASSES;     // N_NODES*N_CLASSES

    const int TB = 256;

    // zero the accumulators (every call: deterministic, capture-safe)
    gcn_zero_kernel<<<(2 * N_NODES + TB - 1) / TB, TB, 0, stream>>>(degO, 2 * N_NODES);
    gcn_zero_kernel<<<(N_NODES * HIDDEN + TB - 1) / TB, TB, 0, stream>>>(agg1, N_NODES * HIDDEN);
    gcn_zero_kernel<<<(N_NODES * N_CLASSES + TB - 1) / TB, TB, 0, stream>>>(agg2, N_NODES * N_CLASSES);

    // degrees -> D^{-1/2}
    gcn_degree_kernel<<<(N_EDGES + TB - 1) / TB, TB, 0, stream>>>(src, dst, degO, degI);
    gcn_isqrt_kernel<<<(N_NODES + TB - 1) / TB, TB, 0, stream>>>(degO, degI);

    // layer 1: h = (X * degO) @ W1   (WMMA f32; 3125 row blocks of 16)
    gcn_wmma_gemm<IN_FEATS, HIDDEN, HIDDEN / 16>
        <<<N_NODES / 16, 32, 0, stream>>>(x, W1, degO, h);
    // agg1 += h[src] scattered to dst
    gcn_scatter128_kernel<<<(N_EDGES * 32) / TB, TB, 0, stream>>>(h, src, dst, agg1);
    // h1 = agg1 * degI + b1 (in place)
    gcn_normbias_kernel<<<(N_NODES * HIDDEN + TB - 1) / TB, TB, 0, stream>>>(agg1, degI, b1);

    // layer 2: h2 = (h1 * degO) @ W2  (40 cols -> 3 padded tiles)
    gcn_wmma_gemm<HIDDEN, N_CLASSES, 3>
        <<<N_NODES / 16, 32, 0, stream>>>(agg1, W2, degO, h2);
    gcn_scatter40_kernel<<<(N_EDGES * N_CLASSES + TB - 1) / TB, TB, 0, stream>>>(h2, src, dst, agg2);

    // normalize + bias + log_softmax
    gcn_lsm_kernel<<<N_NODES / 8, TB, 0, stream>>>(agg2, degI, b2, out);
}